// AttentionRNN_25950192402644
// MI455X (gfx1250) — compile-verified
//
#include <hip/hip_runtime.h>
#include <hip/hip_bf16.h>
#include <stdint.h>

// ---------------------------------------------------------------------------
// AttentionRNN on gfx1250: per-timestep WMMA GEMMs + attention + GRU gate.
// B=1024, T=256, U=256. fp32 WMMA (V_WMMA_F32_16X16X4_F32) to match the fp32
// reference. Weight panels staged into LDS via the Tensor Data Mover
// (tensor_load_to_lds + s_wait_tensorcnt) when the builtin is available.
// ---------------------------------------------------------------------------

typedef __attribute__((ext_vector_type(2))) float v2f;
typedef __attribute__((ext_vector_type(8))) float v8f;
typedef __attribute__((ext_vector_type(4))) unsigned int u32x4;
typedef __attribute__((ext_vector_type(4))) int i32x4;
typedef __attribute__((ext_vector_type(8))) int i32x8;

#define BN 1024
#define TN 256
#define UN 256
#define KDIM 256

// ---------------------------------------------------------------------------
// C[M x ldc] = A[M x K] @ [W1 | W2]   (first n1 columns from W1, rest from W2)
// One 16x16 tile per wave, 4 waves (64 rows) per block. The block's 16-column
// weight panel (256x16 f32 = 16KB) is DMA'd into LDS by the TDM (strided 2D
// tile, row stride ldb) and shared by all 4 waves.
// ---------------------------------------------------------------------------
__global__ void __launch_bounds__(128)
wmma_gemm_dual(const float* __restrict__ A, int lda,
               const float* __restrict__ W1, int ldb1, int n1,
               const float* __restrict__ W2, int ldb2,
               float* __restrict__ C, int ldc)
{
    __shared__ float wlds[KDIM * 16];

    const int tid = threadIdx.x;
    const int n0  = blockIdx.x * 16;

    const float* W;
    int ldb, col;
    if (n0 < n1) { W = W1; ldb = ldb1; col = n0; }
    else         { W = W2; ldb = ldb2; col = n0 - n1; }

    const int wave = tid >> 5;
    const int lane = tid & 31;
    const int lm   = lane & 15;   // M (A) / N (B,C) position
    const int hi   = lane >> 4;   // half-wave select

    const int m0 = (blockIdx.y * 4 + wave) * 16;
    const float* Arow = A + (size_t)(m0 + lm) * lda + 2 * hi;

    // Prefetch this lane's A row (1KB) while the weight panel is staged.
#pragma unroll
    for (int pb = 0; pb < KDIM; pb += 32)
        __builtin_prefetch(Arow + pb, 0, 0);

#if __has_builtin(__builtin_amdgcn_tensor_load_to_lds)
    // ---- Tensor Data Mover: one instruction stages the whole 256x16 panel.
    if (wave == 0) {
        const float* Wtile = W + col;                       // tile start
        const uint64_t ga  = (uint64_t)(uintptr_t)Wtile;    // global address
        const uint32_t la  =
            (uint32_t)(size_t)(__attribute__((address_space(3))) float*)wlds;

        u32x4 g0 = {};
        g0[0] = 1u;                                         // count=1, user desc
        g0[1] = la;                                         // lds_addr
        g0[2] = (uint32_t)ga;                               // global_addr[31:0]
        g0[3] = ((uint32_t)(ga >> 32) & 0x01FFFFFFu)        // global_addr[56:32]
              | (2u << 30);                                 // type = 2 ("image")

        const uint32_t td0 = (uint32_t)ldb;                 // tensor_dim0
        const uint32_t td1 = (uint32_t)KDIM;                // tensor_dim1
        const uint32_t s0  = (uint32_t)ldb;                 // tensor_dim0_stride
        i32x8 g1 = {};
        g1[0] = (int)(2u << 16);                            // data_size=4B
        g1[1] = (int)((td0 & 0xFFFFu) << 16);               // tdim0[15:0]
        g1[2] = (int)(((td0 >> 16) & 0xFFFFu)               // tdim0[31:16]
              | ((td1 & 0xFFFFu) << 16));                   // tdim1[15:0]
        g1[3] = (int)(((td1 >> 16) & 0xFFFFu)               // tdim1[31:16]
              | (16u << 16));                               // tile_dim0 = 16
        g1[4] = (int)(uint32_t)KDIM;                        // tile_dim1 = 256
        g1[5] = (int)s0;                                    // stride0[31:0]
        g1[6] = (int)((s0 >> 16) >> 16);                    // stride0[47:32]
        g1[7] = 0;

        i32x4 z4 = {};
#if __has_include(<hip/amd_detail/amd_gfx1250_TDM.h>)
        i32x8 z8 = {};
        __builtin_amdgcn_tensor_load_to_lds(g0, g1, z4, z4, z8, 0);
#else
        __builtin_amdgcn_tensor_load_to_lds(g0, g1, z4, z4, 0);
#endif
#if __has_builtin(__builtin_amdgcn_s_wait_tensorcnt)
        __builtin_amdgcn_s_wait_tensorcnt(0);
#else
        asm volatile("s_wait_tensorcnt 0x0" ::: "memory");
#endif
    }
#else
    // ---- Fallback: cooperative global->LDS staging.
    for (int i = tid; i < KDIM * 16; i += 128) {
        const int k = i >> 4;
        const int c = i & 15;
        wlds[i] = W[(size_t)k * ldb + col + c];
    }
#endif
    __syncthreads();

    v8f acc = {};
#pragma unroll 8
    for (int k = 0; k < KDIM; k += 4) {
        // A frag: lane lm holds A[m0+lm][k + r + 2*hi] in vgpr r  -> one b64
        v2f a = *(const v2f*)(Arow + k);
        // B frag: lane lm holds W[k + r + 2*hi][n0+lm] in vgpr r
        v2f b;
        b.x = wlds[(k + 2 * hi) * 16 + lm];
        b.y = wlds[(k + 1 + 2 * hi) * 16 + lm];
        acc = __builtin_amdgcn_wmma_f32_16x16x4_f32(
            /*neg_a=*/false, a, /*neg_b=*/false, b,
            /*c_mod=*/(short)0, acc, /*reuse_a=*/false, /*reuse_b=*/false);
    }

    // C frag: vgpr r -> row m0 + r + 8*hi, col n0 + lm
    float* Cout = C + (size_t)m0 * ldc + n0 + lm;
#pragma unroll
    for (int r = 0; r < 8; ++r)
        Cout[(size_t)(r + 8 * hi) * ldc] = acc[r];
}

// ---------------------------------------------------------------------------
// Attention: scores_s = sum_u tanh(EX_s + Q + bias) * lenv ; softmax over 3
// slots (x_{t-2}, x_{t-1}, x_t); attention_out = sum_s prob_s * x_s.
// One block per batch row, 256 threads (u). Invalid (t<2 / t<1) slots have
// value rows == 0 and emb == 0, but their score term tanh(q+bias)*lenv still
// participates in the softmax (matches the reference's zero-init carries).
// ---------------------------------------------------------------------------
__global__ void __launch_bounds__(256)
attn_kernel(const float* __restrict__ QGS,     // B x 1024, Q in cols [0,256)
            const float* __restrict__ ring_m2, // EX ring slot for t-2
            const float* __restrict__ ring_m1, // EX ring slot for t-1
            const float* __restrict__ ring_t,  // EX ring slot for t
            const float* __restrict__ X,       // inputs (B,T,U)
            int t,
            const float* __restrict__ bias,
            const float* __restrict__ lenv,
            float* __restrict__ ATT)           // B x U
{
    const int b = blockIdx.x;
    const int u = threadIdx.x;
    const size_t o1024 = (size_t)b * 1024 + u;

    const float q  = QGS[o1024];
    const float bi = bias[u];
    const float lv = lenv[u];

    const float e0 = (t >= 2) ? ring_m2[o1024] : 0.0f;
    const float e1 = (t >= 1) ? ring_m1[o1024] : 0.0f;
    const float e2 = ring_t[o1024];

    float s0 = tanhf(e0 + q + bi) * lv;
    float s1 = tanhf(e1 + q + bi) * lv;
    float s2 = tanhf(e2 + q + bi) * lv;

    // wave32 butterfly reduce, then cross-wave via LDS
#pragma unroll
    for (int off = 16; off > 0; off >>= 1) {
        s0 += __shfl_xor(s0, off, 32);
        s1 += __shfl_xor(s1, off, 32);
        s2 += __shfl_xor(s2, off, 32);
    }
    __shared__ float red[8][3];
    const int wave = u >> 5;
    const int lane = u & 31;
    if (lane == 0) { red[wave][0] = s0; red[wave][1] = s1; red[wave][2] = s2; }
    __syncthreads();
    float t0 = 0.f, t1 = 0.f, t2 = 0.f;
#pragma unroll
    for (int w = 0; w < 8; ++w) { t0 += red[w][0]; t1 += red[w][1]; t2 += red[w][2]; }

    // softmax over the 3 slots (redundantly in every thread; 3-wide, cheap)
    const float mx = fmaxf(t0, fmaxf(t1, t2));
    float p0 = expf(t0 - mx), p1 = expf(t1 - mx), p2 = expf(t2 - mx);
    const float inv = 1.0f / (p0 + p1 + p2);
    p0 *= inv; p1 *= inv; p2 *= inv;

    const size_t xbase = (size_t)b * TN * UN + u;
    const float v0 = (t >= 2) ? X[xbase + (size_t)(t - 2) * UN] : 0.0f;
    const float v1 = (t >= 1) ? X[xbase + (size_t)(t - 1) * UN] : 0.0f;
    const float v2 = X[xbase + (size_t)t * UN];

    ATT[(size_t)b * UN + u] = p0 * v0 + p1 * v1 + p2 * v2;
}

// ---------------------------------------------------------------------------
// GRU-style gate: u_p, r_p, cand from GS (QGS cols 256..1023), GE, GN (ring
// cols 256..1023); Snext = (1-u_p)*Scur + u_p*cand.
// ---------------------------------------------------------------------------
__global__ void __launch_bounds__(256)
gate_kernel(const float* __restrict__ QGS,   // B x 1024 (GS in cols 256..1023)
            const float* __restrict__ GE,    // B x 768
            const float* __restrict__ RNG,   // ring slot t (GN in cols 256..1023)
            const float* __restrict__ Scur,  // B x U
            float* __restrict__ Snext)       // B x U (or d_out on last step)
{
    const int b = blockIdx.x;
    const int j = threadIdx.x;
    const size_t o1024 = (size_t)b * 1024;
    const size_t o768  = (size_t)b * 768;

    const float gs0 = QGS[o1024 + 256 + j];
    const float gs1 = QGS[o1024 + 512 + j];
    const float gs2 = QGS[o1024 + 768 + j];
    const float gn0 = RNG[o1024 + 256 + j];
    const float gn1 = RNG[o1024 + 512 + j];
    const float gn2 = RNG[o1024 + 768 + j];
    const float ge0 = GE[o768 + j];
    const float ge1 = GE[o768 + 256 + j];
    const float ge2 = GE[o768 + 512 + j];

    const float up   = 1.0f / (1.0f + expf(-(gs0 + ge0 + gn0)));
    const float rp   = 1.0f / (1.0f + expf(-(gs1 + ge1 + gn1)));
    const float cand = tanhf(rp * gs2 + ge2 + gn2);

    const float st = Scur[(size_t)b * UN + j];
    Snext[(size_t)b * UN + j] = (1.0f - up) * st + up * cand;
}

__global__ void zero_kernel(float* __restrict__ p, int n)
{
    const int i = blockIdx.x * blockDim.x + threadIdx.x;
    if (i < n) p[i] = 0.0f;
}

// ---------------------------------------------------------------------------
extern "C" void kernel_launch(void* const* d_in, const int* in_sizes, int n_in,
                              void* d_out, int out_size, void* d_ws, size_t ws_size,
                              hipStream_t stream)
{
    const float* X    = (const float*)d_in[0]; // (B,T,U)
    const float* We   = (const float*)d_in[1]; // (U,U)
    const float* Ws   = (const float*)d_in[2]; // (U,U)
    const float* bias = (const float*)d_in[3]; // (U,)
    const float* lenv = (const float*)d_in[4]; // (U,1)
    const float* Wrs  = (const float*)d_in[5]; // (U,3U)
    const float* Wemb = (const float*)d_in[6]; // (U,3U)
    const float* Wnew = (const float*)d_in[7]; // (U,3U)
    float* out = (float*)d_out;

    // Carve workspace (all fp32): ~23 MB total.
    char* ws = (char*)d_ws;
    size_t off = 0;
    auto alloc = [&](size_t nfloats) {
        float* p = (float*)(ws + off);
        off += nfloats * sizeof(float);
        return p;
    };
    float* S0   = alloc((size_t)BN * UN);      // state ping
    float* S1   = alloc((size_t)BN * UN);      // state pong
    float* QGS  = alloc((size_t)BN * 1024);    // [query | gs]
    float* GE   = alloc((size_t)BN * 768);     // attention_out @ Wemb
    float* ATT  = alloc((size_t)BN * UN);      // attention_out
    float* ring[3] = { alloc((size_t)BN * 1024),
                       alloc((size_t)BN * 1024),
                       alloc((size_t)BN * 1024) }; // [EX_t | GN_t] ring
    (void)ws_size; (void)in_sizes; (void)n_in; (void)out_size;

    zero_kernel<<<(BN * UN + 255) / 256, 256, 0, stream>>>(S0, BN * UN);

    float* Scur  = S0;
    float* Snext = S1;
    const dim3 blk(128);

    for (int t = 0; t < TN; ++t) {
        // [query | gs] = state @ [Ws | Wrs]       (1024 x 1024, K=256)
        wmma_gemm_dual<<<dim3(64, 16), blk, 0, stream>>>(
            Scur, UN, Ws, UN, 256, Wrs, 768, QGS, 1024);

        // ring[t%3] = [x_t @ We | x_t @ Wnew]     (1024 x 1024, K=256)
        wmma_gemm_dual<<<dim3(64, 16), blk, 0, stream>>>(
            X + (size_t)t * UN, TN * UN, We, UN, 256, Wnew, 768, ring[t % 3], 1024);

        // attention scores / softmax / weighted value
        attn_kernel<<<BN, 256, 0, stream>>>(
            QGS, ring[(t + 1) % 3], ring[(t + 2) % 3], ring[t % 3],
            X, t, bias, lenv, ATT);

        // ge = attention_out @ Wemb               (1024 x 768, K=256)
        wmma_gemm_dual<<<dim3(48, 16), blk, 0, stream>>>(
            ATT, UN, Wemb, 768, 768, (const float*)nullptr, 0, GE, 768);

        // gate + state update (last step writes straight to d_out)
        float* dst = (t == TN - 1) ? out : Snext;
        gate_kernel<<<BN, 256, 0, stream>>>(QGS, GE, ring[t % 3], Scur, dst);

        float* tmp = Scur; Scur = Snext; Snext = tmp;
    }
}